// CRCDLoss_76261439308388
// MI455X (gfx1250) — compile-verified
//
#include <hip/hip_runtime.h>
#include <hip/hip_bf16.h>

typedef __attribute__((ext_vector_type(2))) float v2f;
typedef __attribute__((ext_vector_type(8))) float v8f;

#define FEAT_DIM 128
#define BATCH    64
#define KP1      16385      // NCE_K + 1
#define N_DATA_F 100000.0f
#define NCE_T_INV (1.0f / 0.07f)

// ---------------------------------------------------------------------------
// Kernel A: V = X @ W^T + bias  via V_WMMA_F32_16X16X4_F32 (exact fp32).
// One wave (32 threads) per 16x16 output tile. X: [64,K] row-major,
// W: [128,K] row-major (so B = W^T has B[k,n] = W[n,k], contiguous in k).
//
// f32 16x16x4 operand layout (ISA 7.12.2):
//   A (16x4): lanes 0-15 hold M=lane, {VGPR0,VGPR1} = K{0,1}; lanes 16-31 = K{2,3}
//   B (4x16): lanes 0-15 hold N=lane, rows K{0,1}; lanes 16-31 rows K{2,3}
//   C/D (16x16): VGPR j: lanes 0-15 -> M=j, N=lane; lanes 16-31 -> M=j+8
// ---------------------------------------------------------------------------
__global__ void embed_wmma(const float* __restrict__ X, const float* __restrict__ W,
                           const float* __restrict__ bias, float* __restrict__ V,
                           int K) {
    const int tile = blockIdx.x;          // 32 tiles: 4 (M) x 8 (N)
    const int m0 = (tile >> 3) * 16;
    const int n0 = (tile & 7) * 16;
    const int lane = threadIdx.x;
    const int half = lane >> 4;           // 0: K pair {0,1}, 1: K pair {2,3}
    const int l = lane & 15;

    const float* __restrict__ arow = X + (m0 + l) * K + 2 * half;
    const float* __restrict__ brow = W + (n0 + l) * K + 2 * half;

    v8f c = {};
    for (int k0 = 0; k0 < K; k0 += 4) {
        v2f a = *(const v2f*)(arow + k0);
        v2f b = *(const v2f*)(brow + k0);
        c = __builtin_amdgcn_wmma_f32_16x16x4_f32(
                /*neg_a=*/false, a, /*neg_b=*/false, b,
                /*c_mod=*/(short)0, c, /*reuse_a=*/false, /*reuse_b=*/false);
    }

    const float bn = bias[n0 + l];
#pragma unroll
    for (int j = 0; j < 8; ++j) {
        V[(m0 + j + 8 * half) * FEAT_DIM + (n0 + l)] = c[j] + bn;
    }
}

// ---------------------------------------------------------------------------
// Kernel B: in-place row L2 normalization of V [64,128].
// ---------------------------------------------------------------------------
__global__ void l2norm_kernel(float* __restrict__ V) {
    const int b = blockIdx.x;
    const int t = threadIdx.x;            // 128 threads
    float x = V[b * FEAT_DIM + t];
    __shared__ float sh[FEAT_DIM];
    sh[t] = x * x;
    __syncthreads();
    for (int s = FEAT_DIM / 2; s > 0; s >>= 1) {
        if (t < s) sh[t] += sh[t + s];
        __syncthreads();
    }
    V[b * FEAT_DIM + t] = x / sqrtf(sh[0]);
}

// ---------------------------------------------------------------------------
// Kernel C: the bandwidth-dominant phase. One wave per (b,k) pair:
// gather one 512B row from each bank (float4/lane, fully coalesced),
// dot against v_t / v_s, exp(/T). Also emits per-block Z partial sums.
// ---------------------------------------------------------------------------
__global__ void dots_kernel(const float* __restrict__ vs, const float* __restrict__ vt,
                            const float* __restrict__ mem1, const float* __restrict__ mem2,
                            const int* __restrict__ idx, const int* __restrict__ cidx,
                            float* __restrict__ es, float* __restrict__ et,
                            float* __restrict__ zpart) {
    const long TOT = (long)BATCH * KP1;
    const long gw = (long)blockIdx.x * 8 + (threadIdx.x >> 5);  // global wave id
    const int lane = threadIdx.x & 31;

    float le_s = 0.0f, le_t = 0.0f;
    if (gw < TOT) {
        const int b = (int)(gw / KP1);
        const int k = (int)(gw % KP1);
        const int id = (k == 0) ? idx[b] : cidx[(long)b * KP1 + k];

        const float4 m1 = ((const float4*)(mem1 + (long)id * FEAT_DIM))[lane];
        const float4 m2 = ((const float4*)(mem2 + (long)id * FEAT_DIM))[lane];
        const float4 a  = ((const float4*)(vt + b * FEAT_DIM))[lane];
        const float4 s  = ((const float4*)(vs + b * FEAT_DIM))[lane];

        float dt = m1.x * a.x + m1.y * a.y + m1.z * a.z + m1.w * a.w;
        float ds = m2.x * s.x + m2.y * s.y + m2.z * s.z + m2.w * s.w;
#pragma unroll
        for (int off = 16; off > 0; off >>= 1) {
            dt += __shfl_xor(dt, off, 32);
            ds += __shfl_xor(ds, off, 32);
        }
        if (lane == 0) {
            le_t = __expf(dt * NCE_T_INV);
            le_s = __expf(ds * NCE_T_INV);
            et[gw] = le_t;
            es[gw] = le_s;
        }
    }

    __shared__ float sh0[256], sh1[256];
    sh0[threadIdx.x] = le_s;
    sh1[threadIdx.x] = le_t;
    __syncthreads();
    for (int s = 128; s > 0; s >>= 1) {
        if (threadIdx.x < s) {
            sh0[threadIdx.x] += sh0[threadIdx.x + s];
            sh1[threadIdx.x] += sh1[threadIdx.x + s];
        }
        __syncthreads();
    }
    if (threadIdx.x == 0) {
        zpart[2 * (long)blockIdx.x + 0] = sh0[0];
        zpart[2 * (long)blockIdx.x + 1] = sh1[0];
    }
}

// ---------------------------------------------------------------------------
// Kernel D0: reduce Z partials -> Z_s, Z_t (deterministic fixed-order sums).
// ---------------------------------------------------------------------------
__global__ void zreduce_kernel(const float* __restrict__ zpart, int n,
                               float* __restrict__ zscal) {
    __shared__ float sh0[256], sh1[256];
    float ss = 0.0f, st = 0.0f;
    for (int i = threadIdx.x; i < n; i += 256) {
        ss += zpart[2 * (long)i + 0];
        st += zpart[2 * (long)i + 1];
    }
    sh0[threadIdx.x] = ss;
    sh1[threadIdx.x] = st;
    __syncthreads();
    for (int s = 128; s > 0; s >>= 1) {
        if (threadIdx.x < s) {
            sh0[threadIdx.x] += sh0[threadIdx.x + s];
            sh1[threadIdx.x] += sh1[threadIdx.x + s];
        }
        __syncthreads();
    }
    if (threadIdx.x == 0) {
        const float TOT = (float)BATCH * (float)KP1;
        zscal[0] = sh0[0] / TOT * N_DATA_F;  // Z_s = mean(out_s) * n_data
        zscal[1] = sh1[0] / TOT * N_DATA_F;  // Z_t
    }
}

// ---------------------------------------------------------------------------
// Kernel D1: per-element NCE log terms (both banks), block partial sums.
// ---------------------------------------------------------------------------
__global__ void loss_partial_kernel(const float* __restrict__ es, const float* __restrict__ et,
                                    const float* __restrict__ zscal,
                                    float* __restrict__ lpart) {
    const long TOT = (long)BATCH * KP1;
    const long i = (long)blockIdx.x * 256 + threadIdx.x;
    float term = 0.0f;
    if (i < TOT) {
        const int k = (int)(i % KP1);
        const float mPn = 16384.0f / N_DATA_F;
        const float c = mPn + 1e-7f;
        const float os = es[i] / zscal[0];
        const float ot = et[i] / zscal[1];
        if (k == 0)
            term = logf(os / (os + c)) + logf(ot / (ot + c));  // positives
        else
            term = logf(mPn / (os + c)) + logf(mPn / (ot + c));  // negatives
    }
    __shared__ float sh[256];
    sh[threadIdx.x] = term;
    __syncthreads();
    for (int s = 128; s > 0; s >>= 1) {
        if (threadIdx.x < s) sh[threadIdx.x] += sh[threadIdx.x + s];
        __syncthreads();
    }
    if (threadIdx.x == 0) lpart[blockIdx.x] = sh[0];
}

// ---------------------------------------------------------------------------
// Kernel D2: final deterministic reduction -> loss = -(sum)/B.
// ---------------------------------------------------------------------------
__global__ void loss_final_kernel(const float* __restrict__ lpart, int n,
                                  float* __restrict__ out) {
    __shared__ float sh[256];
    float s0 = 0.0f;
    for (int i = threadIdx.x; i < n; i += 256) s0 += lpart[i];
    sh[threadIdx.x] = s0;
    __syncthreads();
    for (int s = 128; s > 0; s >>= 1) {
        if (threadIdx.x < s) sh[threadIdx.x] += sh[threadIdx.x + s];
        __syncthreads();
    }
    if (threadIdx.x == 0) out[0] = -sh[0] / (float)BATCH;
}

// ---------------------------------------------------------------------------
extern "C" void kernel_launch(void* const* d_in, const int* in_sizes, int n_in,
                              void* d_out, int out_size, void* d_ws, size_t ws_size,
                              hipStream_t stream) {
    const float* f_s  = (const float*)d_in[0];   // [64,1024]
    const float* f_t  = (const float*)d_in[1];   // [64,2048]
    const int*   idx  = (const int*)d_in[2];     // [64]
    const int*   cidx = (const int*)d_in[3];     // [64,16385]
    const float* Ws   = (const float*)d_in[4];   // [128,1024]
    const float* bs   = (const float*)d_in[5];   // [128]
    const float* Wt   = (const float*)d_in[6];   // [128,2048]
    const float* bt   = (const float*)d_in[7];   // [128]
    const float* mem1 = (const float*)d_in[8];   // [100000,128]  (v_t side)
    const float* mem2 = (const float*)d_in[9];   // [100000,128]  (v_s side)

    const long TOT = (long)BATCH * KP1;          // 1,048,640 (b,k) pairs
    const int nblocksC = (int)((TOT + 7) / 8);   // 8 waves / 256-thread block
    const int nblocksL = (int)((TOT + 255) / 256);

    // workspace layout (floats)
    float* ws    = (float*)d_ws;
    float* vs    = ws;                           // 64*128
    float* vt    = vs + BATCH * FEAT_DIM;        // 64*128
    float* es    = vt + BATCH * FEAT_DIM;        // TOT
    float* et    = es + TOT;                     // TOT
    float* zpart = et + TOT;                     // 2*nblocksC
    float* zscal = zpart + 2L * nblocksC;        // 2
    float* lpart = zscal + 2;                    // nblocksL

    embed_wmma<<<32, 32, 0, stream>>>(f_s, Ws, bs, vs, 1024);
    embed_wmma<<<32, 32, 0, stream>>>(f_t, Wt, bt, vt, 2048);
    l2norm_kernel<<<BATCH, FEAT_DIM, 0, stream>>>(vs);
    l2norm_kernel<<<BATCH, FEAT_DIM, 0, stream>>>(vt);
    dots_kernel<<<nblocksC, 256, 0, stream>>>(vs, vt, mem1, mem2, idx, cidx, es, et, zpart);
    zreduce_kernel<<<1, 256, 0, stream>>>(zpart, nblocksC, zscal);
    loss_partial_kernel<<<nblocksL, 256, 0, stream>>>(es, et, zscal, lpart);
    loss_final_kernel<<<1, 256, 0, stream>>>(lpart, nblocksL, (float*)d_out);
}